// SegmentPooler_84112639525064
// MI455X (gfx1250) — compile-verified
//
#include <hip/hip_runtime.h>
#include <hip/hip_bf16.h>

typedef __attribute__((ext_vector_type(2))) float v2f;
typedef __attribute__((ext_vector_type(8))) float v8f;

#define NSEG 16

// ---------------------------------------------------------------------------
// Kernel 1: per-batch valid length + integer segment bounds (exact vs ref),
// writes start/end/mask ints to workspace and seg_mask floats to output tail.
// ---------------------------------------------------------------------------
__global__ __launch_bounds__(256) void seg_bounds_kernel(
    const int* __restrict__ attn_mask, int L,
    int* __restrict__ ws, float* __restrict__ segmask_out, int nbs) {
  const int b = blockIdx.x;
  const int t = threadIdx.x;
  __shared__ int red[256];

  int sum = 0;
  for (int i = t; i < L; i += 256) sum += attn_mask[(size_t)b * L + i];
  red[t] = sum;
  __syncthreads();
#pragma unroll
  for (int ofs = 128; ofs > 0; ofs >>= 1) {
    if (t < ofs) red[t] += red[t + ofs];
    __syncthreads();
  }
  const int valid = red[0];

  if (t < NSEG) {
    const int s = t;
    // floor((s/16)*valid) is exact in f32 for these ranges -> integer form.
    int start = (s * valid) >> 4;
    int end   = ((s + 1) * valid) >> 4;
    if (s == NSEG - 1) end = valid;
    const int m = (start < valid) ? 1 : 0;
    if (end <= start) {
      int e = start + 1;
      if (e > valid) e = valid;
      end = e;
    }
    const int idx = b * NSEG + s;
    ws[idx]            = start;
    ws[nbs + idx]      = end;
    ws[2 * nbs + idx]  = m;
    segmask_out[idx]   = m ? 1.0f : 0.0f;
  }
}

// ---------------------------------------------------------------------------
// Kernel 2: segment mean via V_WMMA_F32_16X16X4_F32.
//   A = ones(16x4)  ->  D[m][n] = C[m][n] + sum_k B[k][n]
//   B[k][n] = hidden[row r+k][col h0+n]   (B 4x16 f32: 2 VGPRs,
//     VGPR0 = K0 lanes0-15 / K2 lanes16-31, VGPR1 = K1 / K3)
// CH > 0 bakes H in at compile time so all row loads become immediate-offset
// global_load_b32 off one per-lane base pointer (one 64-bit bump / 16 rows).
// ---------------------------------------------------------------------------
template <int CH>
__global__ __launch_bounds__(128) void seg_pool_wmma_kernel(
    const float* __restrict__ hs, const int* __restrict__ ws,
    float* __restrict__ out, int L, int Hrt, int nbs) {
  const int H    = (CH > 0) ? CH : Hrt;
  const int bs   = blockIdx.x;        // b*16 + s
  const int b    = bs >> 4;
  const int wave = threadIdx.x >> 5;  // wave32
  const int lane = threadIdx.x & 31;
  const int h0   = blockIdx.y * 64 + wave * 16;
  const int col  = h0 + (lane & 15);
  const int rowoff = (lane >> 4) << 1;  // 0 for lanes 0-15, 2 for lanes 16-31

  const int start = ws[bs];
  const int end   = ws[nbs + bs];
  const int m     = ws[2 * nbs + bs];

  float* o = out + (size_t)bs * H + h0;
  if (!m) {                       // uniform per block: EXEC stays full
    if (lane < 16) o[lane] = 0.0f;
    return;
  }

  const v2f a = {1.0f, 1.0f};     // ones A-matrix (16x4 f32 = 2 VGPRs/lane)
  v8f c0 = {}, c1 = {}, c2 = {}, c3 = {};

  // Per-lane base pointer for row (start + rowoff), column col.
  const float* p = hs + (size_t)b * L * H + (size_t)(start + rowoff) * H + col;

  int r = start;
  // ---- main loop: 16 rows, all in range, immediate-offset loads ----
  for (; r + 16 <= end; r += 16) {
    const float x0 = p[0];
    const float x1 = p[(size_t)H];
    const float x2 = p[(size_t)4 * H];
    const float x3 = p[(size_t)5 * H];
    const float x4 = p[(size_t)8 * H];
    const float x5 = p[(size_t)9 * H];
    const float x6 = p[(size_t)12 * H];
    const float x7 = p[(size_t)13 * H];
    const v2f b0 = {x0, x1};
    const v2f b1 = {x2, x3};
    const v2f b2 = {x4, x5};
    const v2f b3 = {x6, x7};
    c0 = __builtin_amdgcn_wmma_f32_16x16x4_f32(false, a, false, b0, (short)0, c0, false, false);
    c1 = __builtin_amdgcn_wmma_f32_16x16x4_f32(false, a, false, b1, (short)0, c1, false, false);
    c2 = __builtin_amdgcn_wmma_f32_16x16x4_f32(false, a, false, b2, (short)0, c2, false, false);
    c3 = __builtin_amdgcn_wmma_f32_16x16x4_f32(false, a, false, b3, (short)0, c3, false, false);
    p += (size_t)16 * H;
  }
  // ---- tail: up to 15 rows, per-lane predicated (EXEC restored for WMMA) --
  for (; r < end; r += 8) {
    const int r0 = r + rowoff;
    v2f b0 = {0.0f, 0.0f};
    v2f b1 = {0.0f, 0.0f};
    if (r0     < end) b0.x = p[0];
    if (r0 + 1 < end) b0.y = p[(size_t)H];
    if (r0 + 4 < end) b1.x = p[(size_t)4 * H];
    if (r0 + 5 < end) b1.y = p[(size_t)5 * H];
    c0 = __builtin_amdgcn_wmma_f32_16x16x4_f32(false, a, false, b0, (short)0, c0, false, false);
    c1 = __builtin_amdgcn_wmma_f32_16x16x4_f32(false, a, false, b1, (short)0, c1, false, false);
    p += (size_t)8 * H;
  }

  int cnt = end - start;
  if (cnt < 1) cnt = 1;
  // Row M=0 of C/D lives in VGPR0, lanes 0-15 hold N=0..15.
  const float mean = ((c0[0] + c1[0]) + (c2[0] + c3[0])) / (float)cnt;
  if (lane < 16) o[lane] = mean;
}

extern "C" void kernel_launch(void* const* d_in, const int* in_sizes, int n_in,
                              void* d_out, int out_size, void* d_ws, size_t ws_size,
                              hipStream_t stream) {
  const float* hs = (const float*)d_in[0];
  const int* am   = (const int*)d_in[1];

  // Derive dims: in_sizes[0]=B*L*H, in_sizes[1]=B*L, out=B*S*H + B*S.
  const int BL = in_sizes[1];
  const int H  = in_sizes[0] / BL;
  const int B  = out_size / (NSEG * (H + 1));
  const int L  = BL / B;
  const int nbs = B * NSEG;

  float* out = (float*)d_out;
  float* segmask_out = out + (size_t)nbs * H;
  int* ws = (int*)d_ws;

  seg_bounds_kernel<<<B, 256, 0, stream>>>(am, L, ws, segmask_out, nbs);

  dim3 grid(nbs, H / 64);
  if (H == 1024) {
    seg_pool_wmma_kernel<1024><<<grid, 128, 0, stream>>>(hs, ws, out, L, H, nbs);
  } else {
    seg_pool_wmma_kernel<0><<<grid, 128, 0, stream>>>(hs, ws, out, L, H, nbs);
  }
}